// GATNetwork_26104811225644
// MI455X (gfx1250) — compile-verified
//
#include <hip/hip_runtime.h>
#include <hip/hip_bf16.h>
#include <math.h>

typedef __attribute__((ext_vector_type(2))) float v2f;
typedef __attribute__((ext_vector_type(8))) float v8f;
typedef __attribute__((ext_vector_type(4))) int   v4i;

#define WAVE 32
#define NEG_SLOPE_GAT 0.2f
#define NEG_SLOPE_ACT 0.01f

// ---------------------------------------------------------------------------
// gfx1250 async global->LDS staging (ASYNCcnt path), with safe fallback
// ---------------------------------------------------------------------------
#if defined(__has_builtin)
#if __has_builtin(__builtin_amdgcn_global_load_async_to_lds_b128) && \
    __has_builtin(__builtin_amdgcn_s_wait_asynccnt)
#define HAVE_ASYNC_LDS 1
#endif
#endif

// ---------------------------------------------------------------------------
// float atomic max via ordered-int trick (works for +/- values, init to -inf)
// ---------------------------------------------------------------------------
__device__ __forceinline__ void atomicMaxFloat(float* addr, float val) {
  if (val >= 0.0f) {
    atomicMax((int*)addr, __float_as_int(val));
  } else {
    atomicMin((unsigned int*)addr, __float_as_uint(val));
  }
}

// ---------------------------------------------------------------------------
// fill
// ---------------------------------------------------------------------------
__global__ void fill_kernel(float* __restrict__ p, long long n, float v) {
  long long i = (long long)blockIdx.x * blockDim.x + threadIdx.x;
  if (i < n) p[i] = v;
}

// ---------------------------------------------------------------------------
// Strip GEMM: C[N,M] = A[N,K] @ B[K,M], K,M compile-time (K=128, M in {64,128}).
// One wave32 per 16-row strip; the wave computes ALL M/16 column tiles so the
// A strip is read exactly once. B is staged block-wide into LDS (async copy
// on gfx1250), and B fragments come from ds_load.
// V_WMMA_F32_16X16X4_F32 fragments (ISA 7.12.2):
//   A 16x4: lanes 0-15 M=lane (K0,K1 in 2 VGPRs); lanes 16-31 K2,K3.
//   B 4x16: lanes 0-15 N=lane (K0,K1);            lanes 16-31 K2,K3.
//   C 16x16: VGPR i -> (M = i + 8*(lane>>4), N = lane&15).
// ---------------------------------------------------------------------------
template <int K, int M>
__global__ void __launch_bounds__(256)
gemm_wmma_strip(const float* __restrict__ A, const float* __restrict__ B,
                float* __restrict__ C, int N) {
  constexpr int CT = M / 16;              // column tiles per strip
  __shared__ float Bs[K * M];

  // cooperative stage of B into LDS: K*M/4 float4 chunks
  constexpr int TOT4 = (K * M) / 4;
  for (int i = threadIdx.x; i < TOT4; i += blockDim.x) {
#if HAVE_ASYNC_LDS
    __builtin_amdgcn_global_load_async_to_lds_b128(
        (__attribute__((address_space(1))) v4i*)(B + 4 * i),
        (__attribute__((address_space(3))) v4i*)(&Bs[4 * i]),
        /*offset=*/0, /*cpol=*/0);
#else
    ((float4*)Bs)[i] = ((const float4*)B)[i];
#endif
  }
#if HAVE_ASYNC_LDS
  __builtin_amdgcn_s_wait_asynccnt(0);
#endif
  __syncthreads();

  int strip = (blockIdx.x * blockDim.x + threadIdx.x) / WAVE;
  int lane  = threadIdx.x & (WAVE - 1);
  int m0 = strip * 16;
  if (m0 >= N) return;                    // wave-uniform exit (EXEC stays full)

  int mr   = lane & 15;                   // M for A-frag, N for B/C-frag
  int half = lane >> 4;                   // selects K pair (K0K1 vs K2K3)

  const float* arow = A + (size_t)(m0 + mr) * K + 2 * half;

  v8f acc[CT];
#pragma unroll
  for (int ct = 0; ct < CT; ++ct) acc[ct] = (v8f){};

  for (int k = 0; k < K; k += 4) {
    v2f a;
    a.x = arow[k];
    a.y = arow[k + 1];
    const float* brow0 = Bs + (k + 2 * half) * M + mr;
#pragma unroll
    for (int ct = 0; ct < CT; ++ct) {
      v2f b;
      b.x = brow0[ct * 16];
      b.y = brow0[ct * 16 + M];
      acc[ct] = __builtin_amdgcn_wmma_f32_16x16x4_f32(
          false, a, false, b, (short)0, acc[ct], false, false);
    }
  }

#pragma unroll
  for (int ct = 0; ct < CT; ++ct) {
#pragma unroll
    for (int i = 0; i < 8; ++i) {
      C[(size_t)(m0 + i + 8 * half) * M + ct * 16 + mr] = acc[ct][i];
    }
  }
}

// Generic fallback (round-0 style), used only if runtime dims ever differ.
__global__ void gemm_wmma_generic(const float* __restrict__ A,
                                  const float* __restrict__ B,
                                  float* __restrict__ C,
                                  int N, int K, int M) {
  int gwave = (blockIdx.x * blockDim.x + threadIdx.x) / WAVE;
  int lane  = threadIdx.x & (WAVE - 1);
  int colTiles = M >> 4;
  int rowTile = gwave / colTiles;
  int colTile = gwave % colTiles;
  if (rowTile * 16 >= N) return;
  int m0 = rowTile * 16, n0 = colTile * 16;
  int mr = lane & 15, half = lane >> 4;
  const float* arow = A + (size_t)(m0 + mr) * K + 2 * half;
  const float* bcol = B + (size_t)(2 * half) * M + n0 + mr;
  v8f acc = {};
  for (int k = 0; k < K; k += 4) {
    v2f a, b;
    a.x = arow[k];
    a.y = arow[k + 1];
    b.x = bcol[(size_t)k * M];
    b.y = bcol[(size_t)(k + 1) * M];
    acc = __builtin_amdgcn_wmma_f32_16x16x4_f32(false, a, false, b, (short)0,
                                                acc, false, false);
  }
#pragma unroll
  for (int i = 0; i < 8; ++i)
    C[(size_t)(m0 + i + 8 * half) * M + n0 + mr] = acc[i];
}

// ---------------------------------------------------------------------------
// a_src[n] = xl[n,:] . att_src ; a_dst[n] = xl[n,:] . att_dst (one wave/node)
// float4 lane loads: F=128 -> one b128 per lane.
// ---------------------------------------------------------------------------
__global__ void att_scores(const float* __restrict__ xl,
                           const float* __restrict__ att_s,
                           const float* __restrict__ att_d,
                           float* __restrict__ a_src,
                           float* __restrict__ a_dst, int N, int F) {
  int node = (blockIdx.x * blockDim.x + threadIdx.x) / WAVE;
  int lane = threadIdx.x & (WAVE - 1);
  if (node >= N) return;
  const float4* row = (const float4*)(xl + (size_t)node * F);
  const float4* as4 = (const float4*)att_s;
  const float4* ad4 = (const float4*)att_d;
  int nv = F >> 2;
  float ss = 0.f, sd = 0.f;
  for (int i = lane; i < nv; i += WAVE) {
    float4 v = row[i], s4 = as4[i], d4 = ad4[i];
    ss += v.x * s4.x + v.y * s4.y + v.z * s4.z + v.w * s4.w;
    sd += v.x * d4.x + v.y * d4.y + v.z * d4.z + v.w * d4.w;
  }
#pragma unroll
  for (int off = 16; off > 0; off >>= 1) {
    ss += __shfl_xor(ss, off, WAVE);
    sd += __shfl_xor(sd, off, WAVE);
  }
  if (lane == 0) { a_src[node] = ss; a_dst[node] = sd; }
}

// ---------------------------------------------------------------------------
// pass 1: per-dst max of leaky_relu(a_src[src]+a_dst[dst]); self-loops appended
// ---------------------------------------------------------------------------
__global__ void edge_max(const int* __restrict__ src, const int* __restrict__ dst,
                         int E, int N,
                         const float* __restrict__ a_src,
                         const float* __restrict__ a_dst,
                         float* __restrict__ m) {
  int e = blockIdx.x * blockDim.x + threadIdx.x;
  int Etot = E + N;
  if (e >= Etot) return;
  int s = (e < E) ? src[e] : (e - E);
  int d = (e < E) ? dst[e] : (e - E);
  float v = a_src[s] + a_dst[d];
  v = (v > 0.f) ? v : NEG_SLOPE_GAT * v;
  atomicMaxFloat(&m[d], v);
}

// ---------------------------------------------------------------------------
// pass 2: ex[e] = exp(e - m[dst]); s[dst] += ex[e]
// ---------------------------------------------------------------------------
__global__ void edge_expsum(const int* __restrict__ src, const int* __restrict__ dst,
                            int E, int N,
                            const float* __restrict__ a_src,
                            const float* __restrict__ a_dst,
                            const float* __restrict__ m,
                            float* __restrict__ exb,
                            float* __restrict__ sbuf) {
  int e = blockIdx.x * blockDim.x + threadIdx.x;
  int Etot = E + N;
  if (e >= Etot) return;
  int s = (e < E) ? src[e] : (e - E);
  int d = (e < E) ? dst[e] : (e - E);
  float v = a_src[s] + a_dst[d];
  v = (v > 0.f) ? v : NEG_SLOPE_GAT * v;
  float ex = expf(v - m[d]);
  exb[e] = ex;
  atomicAdd(&sbuf[d], ex);
}

// ---------------------------------------------------------------------------
// pass 3: agg[dst,:] += alpha * xl[src,:]
// one wave32 per edge; float4 gather (b128), per-float f32 atomics to L2.
// ---------------------------------------------------------------------------
__global__ void edge_agg(const int* __restrict__ src, const int* __restrict__ dst,
                         int E, int N,
                         const float* __restrict__ exb,
                         const float* __restrict__ sbuf,
                         const float* __restrict__ xl,
                         float* __restrict__ agg, int F) {
  int e = (blockIdx.x * blockDim.x + threadIdx.x) / WAVE;
  int lane = threadIdx.x & (WAVE - 1);
  int Etot = E + N;
  if (e >= Etot) return;
  int s = (e < E) ? src[e] : (e - E);
  int d = (e < E) ? dst[e] : (e - E);
  float alpha = exb[e] / (sbuf[d] + 1e-16f);
  const float4* x4 = (const float4*)(xl + (size_t)s * F);
  float* orow = agg + (size_t)d * F;
  int nv = F >> 2;
  for (int i = lane; i < nv; i += WAVE) {
    float4 v = x4[i];
    atomicAdd(&orow[4 * i + 0], alpha * v.x);
    atomicAdd(&orow[4 * i + 1], alpha * v.y);
    atomicAdd(&orow[4 * i + 2], alpha * v.z);
    atomicAdd(&orow[4 * i + 3], alpha * v.w);
  }
}

// ---------------------------------------------------------------------------
// out[i] = in[i] + b[i%F], optional leaky relu
// ---------------------------------------------------------------------------
__global__ void bias_act(const float* __restrict__ in, const float* __restrict__ b,
                         float* __restrict__ out, long long total, int F,
                         float slope, int do_act) {
  long long i = (long long)blockIdx.x * blockDim.x + threadIdx.x;
  if (i >= total) return;
  float v = in[i] + b[(int)(i % F)];
  if (do_act) v = (v > 0.f) ? v : slope * v;
  out[i] = v;
}

// ---------------------------------------------------------------------------
// column (axis-0) softmax reductions over z[N,C], C <= 128, blockDim == 256
// ---------------------------------------------------------------------------
__global__ void col_max(const float* __restrict__ z, int N, int C,
                        float* __restrict__ colmax) {
  __shared__ float red[256];
  int c = threadIdx.x % C;
  int rpb = 256 / C;
  int rib = threadIdx.x / C;
  float mx = -INFINITY;
  for (long long r = (long long)blockIdx.x * rpb + rib; r < N;
       r += (long long)gridDim.x * rpb)
    mx = fmaxf(mx, z[r * C + c]);
  red[threadIdx.x] = mx;
  __syncthreads();
  for (int off = 128; off >= C; off >>= 1) {
    if ((int)threadIdx.x < off)
      red[threadIdx.x] = fmaxf(red[threadIdx.x], red[threadIdx.x + off]);
    __syncthreads();
  }
  if ((int)threadIdx.x < C) atomicMaxFloat(&colmax[c], red[threadIdx.x]);
}

__global__ void col_sum(const float* __restrict__ z, int N, int C,
                        const float* __restrict__ colmax,
                        float* __restrict__ colsum) {
  __shared__ float red[256];
  int c = threadIdx.x % C;
  int rpb = 256 / C;
  int rib = threadIdx.x / C;
  float cm = colmax[c];
  float sm = 0.f;
  for (long long r = (long long)blockIdx.x * rpb + rib; r < N;
       r += (long long)gridDim.x * rpb)
    sm += expf(z[r * C + c] - cm);
  red[threadIdx.x] = sm;
  __syncthreads();
  for (int off = 128; off >= C; off >>= 1) {
    if ((int)threadIdx.x < off)
      red[threadIdx.x] += red[threadIdx.x + off];
    __syncthreads();
  }
  if ((int)threadIdx.x < C) atomicAdd(&colsum[c], red[threadIdx.x]);
}

__global__ void softmax_norm(float* __restrict__ z, long long total, int C,
                             const float* __restrict__ colmax,
                             const float* __restrict__ colsum) {
  long long i = (long long)blockIdx.x * blockDim.x + threadIdx.x;
  if (i >= total) return;
  int c = (int)(i % C);
  z[i] = expf(z[i] - colmax[c]) / colsum[c];
}

// ---------------------------------------------------------------------------
// launch
// ---------------------------------------------------------------------------
static inline int cdiv(long long a, long long b) { return (int)((a + b - 1) / b); }

static void launch_gemm(const float* A, const float* B, float* C,
                        int N, int K, int M, hipStream_t stream) {
  const int TB = 256;
  if (K == 128 && M == 128) {
    long long waves = (N + 15) / 16;   // one wave per strip
    gemm_wmma_strip<128, 128><<<cdiv(waves * WAVE, TB), TB, 0, stream>>>(A, B, C, N);
  } else if (K == 128 && M == 64) {
    long long waves = (N + 15) / 16;
    gemm_wmma_strip<128, 64><<<cdiv(waves * WAVE, TB), TB, 0, stream>>>(A, B, C, N);
  } else {
    long long waves = (long long)((N + 15) / 16) * (M / 16);
    gemm_wmma_generic<<<cdiv(waves * WAVE, TB), TB, 0, stream>>>(A, B, C, N, K, M);
  }
}

extern "C" void kernel_launch(void* const* d_in, const int* in_sizes, int n_in,
                              void* d_out, int out_size, void* d_ws, size_t ws_size,
                              hipStream_t stream) {
  const float* x     = (const float*)d_in[0];
  const int*   eidx  = (const int*)  d_in[1];
  const float* W1    = (const float*)d_in[2];
  const float* atts1 = (const float*)d_in[3];
  const float* attd1 = (const float*)d_in[4];
  const float* b1    = (const float*)d_in[5];
  const float* W2    = (const float*)d_in[6];
  const float* atts2 = (const float*)d_in[7];
  const float* attd2 = (const float*)d_in[8];
  const float* b2    = (const float*)d_in[9];

  const int h1  = in_sizes[3];           // 128
  const int h2  = in_sizes[7];           // 64
  const int din = in_sizes[2] / h1;      // 128
  const int N   = in_sizes[0] / din;     // 100000
  const int E   = in_sizes[1] / 2;       // 1600000
  const int Etot = E + N;
  const int* srcIdx = eidx;
  const int* dstIdx = eidx + E;

  // workspace carve (floats)
  float* ws    = (float*)d_ws;
  float* xl    = ws;                           // N*h1 (layer2 reuses first N*h2)
  float* aggh  = xl   + (size_t)N * h1;        // N*h1 (agg1 -> h ; layer2 agg2)
  float* a_s   = aggh + (size_t)N * h1;        // N
  float* a_d   = a_s  + N;                     // N
  float* mbuf  = a_d  + N;                     // N
  float* sbuf  = mbuf + N;                     // N
  float* exb   = sbuf + N;                     // Etot
  float* cmax  = exb  + Etot;                  // h2
  float* csum  = cmax + h2;                    // h2
  float* zout  = (float*)d_out;                // N*h2

  const int TB = 256;
  const long long Nh1 = (long long)N * h1;
  const long long Nh2 = (long long)N * h2;

  // ---------------- layer 1 ----------------
  launch_gemm(x, W1, xl, N, din, h1, stream);
  att_scores<<<cdiv((long long)N * WAVE, TB), TB, 0, stream>>>(xl, atts1, attd1, a_s, a_d, N, h1);
  fill_kernel<<<cdiv(N, TB), TB, 0, stream>>>(mbuf, N, -INFINITY);
  fill_kernel<<<cdiv(N, TB), TB, 0, stream>>>(sbuf, N, 0.f);
  edge_max   <<<cdiv(Etot, TB), TB, 0, stream>>>(srcIdx, dstIdx, E, N, a_s, a_d, mbuf);
  edge_expsum<<<cdiv(Etot, TB), TB, 0, stream>>>(srcIdx, dstIdx, E, N, a_s, a_d, mbuf, exb, sbuf);
  fill_kernel<<<cdiv(Nh1, TB), TB, 0, stream>>>(aggh, Nh1, 0.f);
  edge_agg   <<<cdiv((long long)Etot * WAVE, TB), TB, 0, stream>>>(srcIdx, dstIdx, E, N, exb, sbuf, xl, aggh, h1);
  bias_act   <<<cdiv(Nh1, TB), TB, 0, stream>>>(aggh, b1, aggh, Nh1, h1, NEG_SLOPE_ACT, 1);

  // ---------------- layer 2 ----------------
  launch_gemm(aggh, W2, xl, N, h1, h2, stream);
  att_scores<<<cdiv((long long)N * WAVE, TB), TB, 0, stream>>>(xl, atts2, attd2, a_s, a_d, N, h2);
  fill_kernel<<<cdiv(N, TB), TB, 0, stream>>>(mbuf, N, -INFINITY);
  fill_kernel<<<cdiv(N, TB), TB, 0, stream>>>(sbuf, N, 0.f);
  edge_max   <<<cdiv(Etot, TB), TB, 0, stream>>>(srcIdx, dstIdx, E, N, a_s, a_d, mbuf);
  edge_expsum<<<cdiv(Etot, TB), TB, 0, stream>>>(srcIdx, dstIdx, E, N, a_s, a_d, mbuf, exb, sbuf);
  fill_kernel<<<cdiv(Nh2, TB), TB, 0, stream>>>(aggh, Nh2, 0.f);   // h no longer needed
  edge_agg   <<<cdiv((long long)Etot * WAVE, TB), TB, 0, stream>>>(srcIdx, dstIdx, E, N, exb, sbuf, xl, aggh, h2);
  bias_act   <<<cdiv(Nh2, TB), TB, 0, stream>>>(aggh, b2, zout, Nh2, h2, 0.f, 0);

  // ---------------- axis-0 softmax ----------------
  fill_kernel<<<1, TB, 0, stream>>>(cmax, h2, -INFINITY);
  fill_kernel<<<1, TB, 0, stream>>>(csum, h2, 0.f);
  col_max<<<1024, TB, 0, stream>>>(zout, N, h2, cmax);
  col_sum<<<1024, TB, 0, stream>>>(zout, N, h2, cmax, csum);
  softmax_norm<<<cdiv(Nh2, TB), TB, 0, stream>>>(zout, Nh2, h2, cmax, csum);
}